// GATModel_678604833256
// MI455X (gfx1250) — compile-verified
//
#include <hip/hip_runtime.h>
#include <math.h>

// ---------------------------------------------------------------------------
// GAT (3 layers, heads=1) for MI455X / gfx1250, fp32 end-to-end.
// Dense per-layer GEMM h' = h @ W uses V_WMMA_F32_16X16X4_F32 (fp32 WMMA).
// Graph softmax/aggregation via fp32 global atomics.
// ---------------------------------------------------------------------------

#define N_NODES  65536
#define N_EDGES  524288
#define E_TOT    (N_EDGES + N_NODES)   // edges + self loops = 589824
#define N_LAYERS 3
#define D        128
#define ED       32
#define NEG_SLOPE 0.2f

typedef float v2f __attribute__((ext_vector_type(2)));
typedef float v8f __attribute__((ext_vector_type(8)));

// ---------------------------------------------------------------------------
// utility: fill a float buffer with a constant
// ---------------------------------------------------------------------------
__global__ void fill_kernel(float* __restrict__ p, size_t n, float v) {
    size_t i = (size_t)blockIdx.x * blockDim.x + threadIdx.x;
    if (i < n) p[i] = v;
}

// float atomic max via signed/unsigned integer atomics (valid with -inf init)
__device__ __forceinline__ void atomicMaxF(float* addr, float v) {
    if (v >= 0.0f) {
        atomicMax((int*)addr, __float_as_int(v));
    } else {
        atomicMin((unsigned int*)addr, __float_as_uint(v));
    }
}

__device__ __forceinline__ float gelu_exact(float x) {
    return 0.5f * x * (1.0f + erff(x * 0.70710678118654752440f));
}

// ---------------------------------------------------------------------------
// Self-loop fill_value='mean': per-dst count and edge_attr sum, then divide.
// ---------------------------------------------------------------------------
__global__ void cnt_kernel(const int* __restrict__ ei, float* __restrict__ cnt) {
    int e = blockIdx.x * blockDim.x + threadIdx.x;
    if (e < N_EDGES) atomicAdd(&cnt[ei[N_EDGES + e]], 1.0f);
}

__global__ void fillsum_kernel(const int* __restrict__ ei,
                               const float* __restrict__ ea,
                               float* __restrict__ fillsum) {
    size_t idx = (size_t)blockIdx.x * blockDim.x + threadIdx.x; // over E*ED
    if (idx >= (size_t)N_EDGES * ED) return;
    int e = (int)(idx >> 5);
    int j = (int)(idx & 31);
    int dst = ei[N_EDGES + e];
    atomicAdd(&fillsum[(size_t)dst * ED + j], ea[idx]);
}

__global__ void filldiv_kernel(float* __restrict__ fill, const float* __restrict__ cnt) {
    size_t idx = (size_t)blockIdx.x * blockDim.x + threadIdx.x; // over N*ED
    if (idx >= (size_t)N_NODES * ED) return;
    float c = cnt[idx >> 5];
    fill[idx] = fill[idx] / fmaxf(c, 1.0f);
}

// ---------------------------------------------------------------------------
// ve[l][j] = sum_d W_edge[l][j][d] * att_edge[l][d]   (3*32 outputs)
// ---------------------------------------------------------------------------
__global__ void ve_kernel(const float* __restrict__ We, const float* __restrict__ ae,
                          float* __restrict__ ve) {
    int t = blockIdx.x * blockDim.x + threadIdx.x;
    if (t >= N_LAYERS * ED) return;
    int l = t / ED, j = t % ED;
    const float* w = We + ((size_t)l * ED + j) * D;
    const float* a = ae + (size_t)l * D;
    float s = 0.0f;
    for (int d = 0; d < D; ++d) s += w[d] * a[d];
    ve[t] = s;
}

// Per-edge edge-attention contribution for all 3 layers at once.
__global__ void eal3_kernel(const float* __restrict__ ea, const float* __restrict__ fill,
                            const float* __restrict__ ve, float* __restrict__ eal3) {
    int e = blockIdx.x * blockDim.x + threadIdx.x;
    if (e >= E_TOT) return;
    const float* p = (e < N_EDGES) ? (ea + (size_t)e * ED)
                                   : (fill + (size_t)(e - N_EDGES) * ED);
    float a0 = 0.0f, a1 = 0.0f, a2 = 0.0f;
    for (int j = 0; j < ED; ++j) {
        float v = p[j];
        a0 += v * ve[j];
        a1 += v * ve[ED + j];
        a2 += v * ve[2 * ED + j];
    }
    eal3[(size_t)e * 3 + 0] = a0;
    eal3[(size_t)e * 3 + 1] = a1;
    eal3[(size_t)e * 3 + 2] = a2;
}

// ---------------------------------------------------------------------------
// WMMA GEMM: C[N_NODES x 128] = A[N_NODES x 128] * B[128 x 128], fp32.
// Block = 256 threads = 8 waves; block b owns rows [16b,16b+16); wave w owns
// cols [16w,16w+16). K-loop in steps of 4 with V_WMMA_F32_16X16X4_F32.
// VGPR layouts per CDNA5 ISA 7.12.2:
//   A 16x4 : lane(M)=0..15 | lanes 16..31 repeat M with K+2; v0=K, v1=K+1
//   B 4x16 : lane(N)=0..15 | lanes 16..31 repeat N with K+2; v0=K, v1=K+1
//   C 16x16: vg g -> M=g (lanes 0..15) / M=g+8 (lanes 16..31), N=lane%16
// ---------------------------------------------------------------------------
__global__ void __launch_bounds__(256)
gemm_wmma_kernel(const float* __restrict__ A, const float* __restrict__ B,
                 float* __restrict__ C) {
    const int lane  = threadIdx.x & 31;
    const int wave  = threadIdx.x >> 5;
    const int half  = lane >> 4;      // 0: K+0/1, 1: K+2/3
    const int l16   = lane & 15;
    const int mbase = blockIdx.x << 4;
    const int nbase = wave << 4;

    v8f acc = {};
    const float* arow = A + (size_t)(mbase + l16) * D;

    #pragma unroll 4
    for (int k = 0; k < D; k += 4) {
        const int kk = k + half * 2;
        v2f a = *(const v2f*)(arow + kk);                 // A[m][kk], A[m][kk+1]
        v2f b;
        b.x = B[(size_t)kk * D + nbase + l16];            // B[kk][n]
        b.y = B[(size_t)(kk + 1) * D + nbase + l16];      // B[kk+1][n]
        acc = __builtin_amdgcn_wmma_f32_16x16x4_f32(
            /*neg_a=*/false, a, /*neg_b=*/false, b,
            /*c_mod=*/(short)0, acc, /*reuse_a=*/false, /*reuse_b=*/false);
    }

    const int mrow = mbase + half * 8;
    #pragma unroll
    for (int g = 0; g < 8; ++g)
        C[(size_t)(mrow + g) * D + nbase + l16] = acc[g];
}

// ---------------------------------------------------------------------------
// alpha_src[i] = Wh[i,:].a_s ; alpha_dst[i] = Wh[i,:].a_d   (wave per node)
// ---------------------------------------------------------------------------
__global__ void alphas_kernel(const float* __restrict__ Wh,
                              const float* __restrict__ a_s,
                              const float* __restrict__ a_d,
                              float* __restrict__ asrc, float* __restrict__ adst) {
    int node = (int)(((size_t)blockIdx.x * blockDim.x + threadIdx.x) >> 5);
    int lane = threadIdx.x & 31;
    if (node >= N_NODES) return;
    float4 h = *(const float4*)(Wh + (size_t)node * D + lane * 4);
    float4 s = *(const float4*)(a_s + lane * 4);
    float4 d = *(const float4*)(a_d + lane * 4);
    float vs = h.x * s.x + h.y * s.y + h.z * s.z + h.w * s.w;
    float vd = h.x * d.x + h.y * d.y + h.z * d.z + h.w * d.w;
    for (int o = 16; o > 0; o >>= 1) {
        vs += __shfl_xor(vs, o, 32);
        vd += __shfl_xor(vd, o, 32);
    }
    if (lane == 0) { asrc[node] = vs; adst[node] = vd; }
}

// ---------------------------------------------------------------------------
// Edge pass 1: logit + leaky relu, store, segment-max over dst.
// ---------------------------------------------------------------------------
__global__ void edge_pass1_kernel(const int* __restrict__ ei,
                                  const float* __restrict__ asrc,
                                  const float* __restrict__ adst,
                                  const float* __restrict__ eal3, int layer,
                                  float* __restrict__ al, float* __restrict__ m) {
    int e = blockIdx.x * blockDim.x + threadIdx.x;
    if (e >= E_TOT) return;
    int src, dst;
    if (e < N_EDGES) { src = ei[e]; dst = ei[N_EDGES + e]; }
    else             { src = dst = e - N_EDGES; }
    float a = asrc[src] + adst[dst] + eal3[(size_t)e * 3 + layer];
    a = (a > 0.0f) ? a : NEG_SLOPE * a;
    al[e] = a;
    atomicMaxF(&m[dst], a);
}

// Edge pass 2: ex = exp(al - m[dst]); den[dst] += ex.
__global__ void edge_pass2_kernel(const int* __restrict__ ei,
                                  float* __restrict__ al,
                                  const float* __restrict__ m,
                                  float* __restrict__ den) {
    int e = blockIdx.x * blockDim.x + threadIdx.x;
    if (e >= E_TOT) return;
    int dst = (e < N_EDGES) ? ei[N_EDGES + e] : (e - N_EDGES);
    float ex = expf(al[e] - m[dst]);
    al[e] = ex;
    atomicAdd(&den[dst], ex);
}

// Edge pass 3: out[dst,:] += (ex/den[dst]) * Wh[src,:]  (wave per edge)
__global__ void edge_pass3_kernel(const int* __restrict__ ei,
                                  const float* __restrict__ al,
                                  const float* __restrict__ den,
                                  const float* __restrict__ Wh,
                                  float* __restrict__ out) {
    int e = (int)(((size_t)blockIdx.x * blockDim.x + threadIdx.x) >> 5);
    int lane = threadIdx.x & 31;
    if (e >= E_TOT) return;
    int src, dst;
    if (e < N_EDGES) { src = ei[e]; dst = ei[N_EDGES + e]; }
    else             { src = dst = e - N_EDGES; }
    float w = al[e] / den[dst];
    float4 h = *(const float4*)(Wh + (size_t)src * D + lane * 4);
    float* o = out + (size_t)dst * D + lane * 4;
    atomicAdd(o + 0, w * h.x);
    atomicAdd(o + 1, w * h.y);
    atomicAdd(o + 2, w * h.z);
    atomicAdd(o + 3, w * h.w);
}

// Epilogue: x = gelu(x + bias[col])  (in place)
__global__ void epilogue_kernel(float* __restrict__ buf, const float* __restrict__ bias) {
    size_t i = (size_t)blockIdx.x * blockDim.x + threadIdx.x;
    if (i >= (size_t)N_NODES * D) return;
    buf[i] = gelu_exact(buf[i] + bias[i & (D - 1)]);
}

// ---------------------------------------------------------------------------
// Host-side orchestration
// ---------------------------------------------------------------------------
static inline size_t align_up(size_t v) { return (v + 255) & ~(size_t)255; }

extern "C" void kernel_launch(void* const* d_in, const int* in_sizes, int n_in,
                              void* d_out, int out_size, void* d_ws, size_t ws_size,
                              hipStream_t stream) {
    const float* x        = (const float*)d_in[0];
    const int*   ei       = (const int*)  d_in[1];   // [2, E]
    const float* ea       = (const float*)d_in[2];   // [E, ED]
    const float* Ws       = (const float*)d_in[3];   // [L, D, D]
    const float* att_src  = (const float*)d_in[4];   // [L, D]
    const float* att_dst  = (const float*)d_in[5];   // [L, D]
    const float* W_edge   = (const float*)d_in[6];   // [L, ED, D]
    const float* att_edge = (const float*)d_in[7];   // [L, D]
    const float* bias     = (const float*)d_in[8];   // [L, D]
    float* out = (float*)d_out;

    // workspace carve-out
    char* ws = (char*)d_ws;
    size_t off = 0;
    auto carve = [&](size_t nfloats) {
        float* p = (float*)(ws + off);
        off += align_up(nfloats * sizeof(float));
        return p;
    };
    float* hA    = carve((size_t)N_NODES * D);   // Wh buffer
    float* hB    = carve((size_t)N_NODES * D);   // ping-pong h buffer
    float* fill  = carve((size_t)N_NODES * ED);  // self-loop mean edge attr
    float* cnt   = carve(N_NODES);
    float* ve    = carve(N_LAYERS * ED);
    float* eal3  = carve((size_t)E_TOT * 3);
    float* al    = carve(E_TOT);
    float* mbuf  = carve(N_NODES);
    float* den   = carve(N_NODES);
    float* asrc  = carve(N_NODES);
    float* adst  = carve(N_NODES);
    (void)ws_size;

    const int T = 256;
    auto blocks = [](size_t n, int t) { return (unsigned)((n + t - 1) / t); };

    // ---- one-time graph preprocessing -------------------------------------
    fill_kernel<<<blocks(N_NODES, T), T, 0, stream>>>(cnt, N_NODES, 0.0f);
    fill_kernel<<<blocks((size_t)N_NODES * ED, T), T, 0, stream>>>(fill, (size_t)N_NODES * ED, 0.0f);
    cnt_kernel<<<blocks(N_EDGES, T), T, 0, stream>>>(ei, cnt);
    fillsum_kernel<<<blocks((size_t)N_EDGES * ED, T), T, 0, stream>>>(ei, ea, fill);
    filldiv_kernel<<<blocks((size_t)N_NODES * ED, T), T, 0, stream>>>(fill, cnt);
    ve_kernel<<<1, N_LAYERS * ED, 0, stream>>>(W_edge, att_edge, ve);
    eal3_kernel<<<blocks(E_TOT, T), T, 0, stream>>>(ea, fill, ve, eal3);

    // ---- layers -----------------------------------------------------------
    // L0: cur=x  -> Wh=hA -> next=hB
    // L1: cur=hB -> Wh=hA -> next=hB   (cur dead after GEMM, alias is safe)
    // L2: cur=hB -> Wh=hA -> next=out
    const float* cur = x;
    for (int l = 0; l < N_LAYERS; ++l) {
        float* Wh   = hA;
        float* next = (l == N_LAYERS - 1) ? out : hB;

        // h' = cur @ W[l]   (WMMA fp32)
        gemm_wmma_kernel<<<N_NODES / 16, 256, 0, stream>>>(
            cur, Ws + (size_t)l * D * D, Wh);

        // per-node attention projections
        alphas_kernel<<<blocks((size_t)N_NODES * 32, T), T, 0, stream>>>(
            Wh, att_src + (size_t)l * D, att_dst + (size_t)l * D, asrc, adst);

        // segment softmax
        fill_kernel<<<blocks(N_NODES, T), T, 0, stream>>>(mbuf, N_NODES, -__builtin_huge_valf());
        fill_kernel<<<blocks(N_NODES, T), T, 0, stream>>>(den, N_NODES, 0.0f);
        fill_kernel<<<blocks((size_t)N_NODES * D, T), T, 0, stream>>>(next, (size_t)N_NODES * D, 0.0f);

        edge_pass1_kernel<<<blocks(E_TOT, T), T, 0, stream>>>(ei, asrc, adst, eal3, l, al, mbuf);
        edge_pass2_kernel<<<blocks(E_TOT, T), T, 0, stream>>>(ei, al, mbuf, den);
        edge_pass3_kernel<<<blocks((size_t)E_TOT * 32, T), T, 0, stream>>>(ei, al, den, Wh, next);

        // bias + GELU
        epilogue_kernel<<<blocks((size_t)N_NODES * D, T), T, 0, stream>>>(
            next, bias + (size_t)l * D);

        cur = next;
    }
}